// DenseGCN1_57458072486027
// MI455X (gfx1250) — compile-verified
//
#include <hip/hip_runtime.h>

typedef float v8f __attribute__((ext_vector_type(8)));
typedef float v2f __attribute__((ext_vector_type(2)));

#define BB 8
#define NN 2048
#define KK 20
#define GG 64
#define NEG_INF (-3.402823466e38f)

// ---------------------------------------------------------------------------
// sq[b,n] = sum_c X[b,c,n]^2
// ---------------------------------------------------------------------------
__global__ void sq_kernel(const float* __restrict__ X, float* __restrict__ sq, int Cin) {
  int t = blockIdx.x * blockDim.x + threadIdx.x;       // over B*N
  int b = t >> 11, n = t & (NN - 1);
  const float* xp = X + (size_t)b * Cin * NN + n;
  float s = 0.f;
  for (int c = 0; c < Cin; ++c) {
    float v = xp[(size_t)c * NN];
    s += v * v;
  }
  sq[t] = s;
}

// ---------------------------------------------------------------------------
// KNN: one workgroup per (b, 16-row strip). Gram strip via V_WMMA_F32_16X16X4_F32
// into LDS, then top-20 per row by iterative shuffle-argmax (16 lanes/row).
// dynamic LDS: dist[16*2048] | As[Cin*16] | sqs[16]
// ---------------------------------------------------------------------------
__global__ void knn_kernel(const float* __restrict__ X, const float* __restrict__ sq,
                           int* __restrict__ idx, int Cin) {
  extern __shared__ float smem[];
  float* dist = smem;                 // 16 * 2048
  float* As   = smem + 16 * NN;       // As[c*16 + m]
  float* sqs  = As + Cin * 16;        // 16 row norms

  int b  = blockIdx.x / (NN / 16);
  int rt = blockIdx.x % (NN / 16);
  int n0 = rt * 16;
  int tid = threadIdx.x;

  // stage the 16 A-rows (points n0..n0+15, all Cin channels)
  for (int i = tid; i < Cin * 16; i += 256) {
    int c = i >> 4, m = i & 15;
    As[i] = X[((size_t)b * Cin + c) * NN + n0 + m];
  }
  if (tid < 16) sqs[tid] = sq[b * NN + n0 + tid];
  __syncthreads();

  int wave = tid >> 5;
  int lane = tid & 31;
  int m16  = lane & 15;
  int h    = lane >> 4;

  // each wave computes 16 contiguous 16-col tiles of the 16x2048 Gram strip
  for (int cti = 0; cti < 16; ++cti) {
    int ct   = wave * 16 + cti;
    int ncol = ct * 16 + m16;
    const float* xb = X + (size_t)b * Cin * NN + ncol;
    v8f acc = {};
    for (int k0 = 0; k0 < Cin; k0 += 4) {
      int c0 = k0 + 2 * h;
      v2f a, bf;
      a.x  = As[c0 * 16 + m16];
      a.y  = As[(c0 + 1) * 16 + m16];
      bf.x = xb[(size_t)c0 * NN];
      bf.y = xb[(size_t)(c0 + 1) * NN];
      acc = __builtin_amdgcn_wmma_f32_16x16x4_f32(false, a, false, bf,
                                                  (short)0, acc, false, false);
    }
    float sqc = sq[b * NN + ncol];
#pragma unroll
    for (int r = 0; r < 8; ++r) {
      int row = r + 8 * h;
      dist[row * NN + ncol] = 2.f * acc[r] - sqs[row] - sqc;  // neg squared distance
    }
  }
  __syncthreads();

  // selection: 16 lanes per row, 20 rounds of argmax with low-index tie-break
  int row = tid >> 4;
  int sub = tid & 15;
  float* drow = dist + row * NN;
  int*   orow = idx + ((size_t)b * NN + n0 + row) * KK;
  for (int it = 0; it < KK; ++it) {
    float bv = NEG_INF;
    int   bi = -1;
    for (int j = sub; j < NN; j += 16) {
      float v = drow[j];
      if (v > bv || (v == bv && (unsigned)j < (unsigned)bi)) { bv = v; bi = j; }
    }
#pragma unroll
    for (int off = 8; off >= 1; off >>= 1) {
      float ov = __shfl_xor(bv, off, 16);
      int   oi = __shfl_xor(bi, off, 16);
      if (ov > bv || (ov == bv && (unsigned)oi < (unsigned)bi)) { bv = ov; bi = oi; }
    }
    if (sub == 0) {
      orow[it]  = bi;
      drow[bi]  = NEG_INF;   // mask winner; lanes of this row share a wave (in-order LDS)
    }
    __builtin_amdgcn_wave_barrier();
  }
}

// ---------------------------------------------------------------------------
// Yi = (W[:, :Cin] - W[:, Cin:]) @ X + bias, Yj = W[:, Cin:] @ X
// stored as (B, N, G). One wave per (b, 16-col tile, 16-row tile of G).
// ---------------------------------------------------------------------------
__global__ void gemm_kernel(const float* __restrict__ X, const float* __restrict__ W,
                            const float* __restrict__ bias,
                            float* __restrict__ Yi, float* __restrict__ Yj, int Cin) {
  int gw   = blockIdx.x * 8 + (threadIdx.x >> 5);  // global wave id: b*512 + nt*4 + mt
  int lane = threadIdx.x & 31;
  int mt = gw & 3;
  int nt = (gw >> 2) & 127;
  int b  = gw >> 9;
  int m16 = lane & 15, h = lane >> 4;
  int o = mt * 16 + m16;
  int n = nt * 16 + m16;
  const float* wrow = W + (size_t)o * (2 * Cin);
  const float* xb   = X + (size_t)b * Cin * NN + n;
  v8f acci = {}, accj = {};
  for (int k0 = 0; k0 < Cin; k0 += 4) {
    int c0 = k0 + 2 * h;
    float w1a = wrow[c0],        w1b = wrow[c0 + 1];
    float w2a = wrow[Cin + c0],  w2b = wrow[Cin + c0 + 1];
    v2f ai, aj, bf;
    ai.x = w1a - w2a; ai.y = w1b - w2b;
    aj.x = w2a;       aj.y = w2b;
    bf.x = xb[(size_t)c0 * NN];
    bf.y = xb[(size_t)(c0 + 1) * NN];
    acci = __builtin_amdgcn_wmma_f32_16x16x4_f32(false, ai, false, bf,
                                                 (short)0, acci, false, false);
    accj = __builtin_amdgcn_wmma_f32_16x16x4_f32(false, aj, false, bf,
                                                 (short)0, accj, false, false);
  }
  float* yi = Yi + ((size_t)b * NN + n) * GG + mt * 16 + 8 * h;
  float* yj = Yj + ((size_t)b * NN + n) * GG + mt * 16 + 8 * h;
#pragma unroll
  for (int r = 0; r < 8; ++r) {
    yi[r] = acci[r] + bias[mt * 16 + 8 * h + r];
    yj[r] = accj[r];
  }
}

// ---------------------------------------------------------------------------
// per-(b,n,o): y_k = yi + yj[idx_k]; track max/min over k; per-channel
// sum/sumsq partials per block (deterministic two-stage reduction).
// 256 threads = 64 channels x 4 points; each block covers 64 points.
// ---------------------------------------------------------------------------
__global__ void aggregate_kernel(const float* __restrict__ Yi, const float* __restrict__ Yj,
                                 const int* __restrict__ idx,
                                 float* __restrict__ Ymax, float* __restrict__ Ymin,
                                 float* __restrict__ partial) {
  int o = threadIdx.x & 63;
  int p = threadIdx.x >> 6;                 // 0..3
  int base = blockIdx.x * 64;               // point id in [0, B*N)
  float s = 0.f, ss = 0.f;
  for (int it = 0; it < 16; ++it) {
    int pt = base + it * 4 + p;
    int b  = pt >> 11;
    float yiv = Yi[(size_t)pt * GG + o];
    const int* ip = idx + (size_t)pt * KK;
    const float* yjb = Yj + (size_t)b * NN * GG + o;
    float mx = NEG_INF, mn = 3.402823466e38f;
    for (int k = 0; k < KK; ++k) {
      float y = yiv + yjb[(size_t)ip[k] * GG];
      mx = fmaxf(mx, y);
      mn = fminf(mn, y);
      s += y;
      ss += y * y;
    }
    Ymax[(size_t)pt * GG + o] = mx;
    Ymin[(size_t)pt * GG + o] = mn;
  }
  __shared__ float red[2][256];
  red[0][threadIdx.x] = s;
  red[1][threadIdx.x] = ss;
  __syncthreads();
  if (p == 0) {
    float ts  = red[0][o] + red[0][64 + o] + red[0][128 + o] + red[0][192 + o];
    float tss = red[1][o] + red[1][64 + o] + red[1][128 + o] + red[1][192 + o];
    partial[(size_t)blockIdx.x * 128 + o]      = ts;
    partial[(size_t)blockIdx.x * 128 + 64 + o] = tss;
  }
}

// ---------------------------------------------------------------------------
// finalize BN affine: scale/shift per channel from partials
// ---------------------------------------------------------------------------
__global__ void stats_kernel(const float* __restrict__ partial, int nblk,
                             const float* __restrict__ gamma, const float* __restrict__ beta,
                             float* __restrict__ scaleArr, float* __restrict__ biasArr) {
  int o = threadIdx.x;   // 64 threads
  float s = 0.f, ss = 0.f;
  for (int i = 0; i < nblk; ++i) {
    s  += partial[(size_t)i * 128 + o];
    ss += partial[(size_t)i * 128 + 64 + o];
  }
  float cnt = (float)(BB * NN * KK);
  float mu  = s / cnt;
  float var = ss / cnt - mu * mu;
  float sc  = gamma[o] * rsqrtf(var + 1e-5f);
  scaleArr[o] = sc;
  biasArr[o]  = beta[o] - mu * sc;
}

// ---------------------------------------------------------------------------
// out[b,o,n] = relu(sc*(sc>=0 ? ymax : ymin) + bi) into channel-major Xnext
// (monotonicity: max_k relu(a*y+c) = relu(a*ysel+c))
// ---------------------------------------------------------------------------
__global__ void finalize_kernel(const float* __restrict__ Ymax, const float* __restrict__ Ymin,
                                const float* __restrict__ scaleArr, const float* __restrict__ biasArr,
                                float* __restrict__ Xn, int Cnext) {
  int t = blockIdx.x * blockDim.x + threadIdx.x;  // over B*64*N, n fastest
  int n = t & (NN - 1);
  int o = (t >> 11) & 63;
  int b = t >> 17;
  size_t pt = (size_t)b * NN + n;
  float sc = scaleArr[o], bi = biasArr[o];
  float y  = (sc >= 0.f) ? Ymax[pt * GG + o] : Ymin[pt * GG + o];
  Xn[((size_t)b * Cnext + o) * NN + n] = fmaxf(sc * y + bi, 0.f);
}

// ---------------------------------------------------------------------------
// dst[b, co+c, n] = src[b, c, n]   (channel-major concat copy)
// ---------------------------------------------------------------------------
__global__ void concat_copy_kernel(const float* __restrict__ src, float* __restrict__ dst,
                                   int Cs, int Cd, int co) {
  size_t t = (size_t)blockIdx.x * blockDim.x + threadIdx.x;  // over B*Cs*N
  int n = (int)(t & (NN - 1));
  size_t cc = t >> 11;
  int c = (int)(cc % Cs);
  int b = (int)(cc / Cs);
  dst[((size_t)b * Cd + co + c) * NN + n] = src[((size_t)b * Cs + c) * NN + n];
}

// ---------------------------------------------------------------------------
extern "C" void kernel_launch(void* const* d_in, const int* in_sizes, int n_in,
                              void* d_out, int out_size, void* d_ws, size_t ws_size,
                              hipStream_t stream) {
  const float* X0  = (const float*)d_in[0];
  const float* W1  = (const float*)d_in[1];
  const float* b1  = (const float*)d_in[2];
  const float* g1  = (const float*)d_in[3];
  const float* be1 = (const float*)d_in[4];
  const float* W2  = (const float*)d_in[5];
  const float* b2  = (const float*)d_in[6];
  const float* g2  = (const float*)d_in[7];
  const float* be2 = (const float*)d_in[8];
  const float* W3  = (const float*)d_in[9];
  const float* b3  = (const float*)d_in[10];
  const float* g3  = (const float*)d_in[11];
  const float* be3 = (const float*)d_in[12];

  float* ws = (float*)d_ws;
  size_t off = 0;
  float* X1      = ws + off; off += (size_t)BB * 128 * NN;   // 2,097,152
  float* X2      = ws + off; off += (size_t)BB * 192 * NN;   // 3,145,728
  float* sqv     = ws + off; off += (size_t)BB * NN;
  int*   idx     = (int*)(ws + off); off += (size_t)BB * NN * KK;
  float* Yi      = ws + off; off += (size_t)BB * NN * GG;
  float* Yj      = ws + off; off += (size_t)BB * NN * GG;
  float* Ymax    = ws + off; off += (size_t)BB * NN * GG;
  float* Ymin    = ws + off; off += (size_t)BB * NN * GG;
  float* partial = ws + off; off += 256 * 128;
  float* scaleA  = ws + off; off += 64;
  float* biasA   = ws + off; off += 64;

  const int nblkAgg = (BB * NN) / 64;            // 256
  const int knnGrid = BB * (NN / 16);            // 1024
  const int gemmGrid = (BB * (NN / 16) * 4) / 8; // 512 blocks of 8 waves

  struct Blk { const float* X; int Cin; const float* W; const float* b;
               const float* g; const float* be; float* Xn; int Cnext; };
  Blk blks[3] = {
    { X0, 64,  W1, b1, g1, be1, X1,             128 },
    { X1, 128, W2, b2, g2, be2, X2,             192 },
    { X2, 192, W3, b3, g3, be3, (float*)d_out,  256 },
  };

  for (int i = 0; i < 3; ++i) {
    const Blk& bl = blks[i];
    int Cin = bl.Cin;
    size_t shmem = (size_t)(16 * NN + Cin * 16 + 16) * sizeof(float);

    sq_kernel<<<(BB * NN) / 256, 256, 0, stream>>>(bl.X, sqv, Cin);
    knn_kernel<<<knnGrid, 256, shmem, stream>>>(bl.X, sqv, idx, Cin);
    gemm_kernel<<<gemmGrid, 256, 0, stream>>>(bl.X, bl.W, bl.b, Yi, Yj, Cin);
    aggregate_kernel<<<nblkAgg, 256, 0, stream>>>(Yi, Yj, idx, Ymax, Ymin, partial);
    stats_kernel<<<1, 64, 0, stream>>>(partial, nblkAgg, bl.g, bl.be, scaleA, biasA);
    finalize_kernel<<<(BB * 64 * NN) / 256, 256, 0, stream>>>(Ymax, Ymin, scaleA, biasA,
                                                              bl.Xn, bl.Cnext);
    concat_copy_kernel<<<((size_t)BB * Cin * NN) / 256, 256, 0, stream>>>(
        bl.X, bl.Xn, Cin, bl.Cnext, 64);
  }
}